// GravityGuidedDebiasModule_38663295599085
// MI455X (gfx1250) — compile-verified
//
#include <hip/hip_runtime.h>
#include <hip/hip_bf16.h>

namespace {
constexpr int kB = 64;
constexpr int kH = 1024;
constexpr int kW = 1024;
constexpr int kN = 256;
constexpr int kRadius = 3;
constexpr int kMaxIters = 20;
constexpr int kStrip = 64;          // columns per LDS strip of the 256x256 pair matrices
constexpr int kStripPad = kStrip + 1; // +1 word: bank = (row+col)%64 -> conflict-free scans
}

typedef float v2f __attribute__((ext_vector_type(2)));
typedef float v8f __attribute__((ext_vector_type(8)));

// 3x3 zero-padded box filter value at (y,x), computed on the fly (border-safe).
__device__ __forceinline__ float smooth_at(const float* __restrict__ img, int y, int x) {
  float s = 0.0f;
#pragma unroll
  for (int ey = -1; ey <= 1; ++ey) {
    int yy = y + ey;
    int yc = min(max(yy, 0), kH - 1);
    bool yok = (yy >= 0) && (yy < kH);
#pragma unroll
    for (int ex = -1; ex <= 1; ++ex) {
      int xx = x + ex;
      int xc = min(max(xx, 0), kW - 1);
      bool ok = yok && (xx >= 0) && (xx < kW);
      s += (ok ? 1.0f : 0.0f) * img[yc * kW + xc];
    }
  }
  return s * (1.0f / 9.0f);
}

// One gravity step for a point with all taps strictly inside the image
// (x,y in [4, 1019]): separable 3x3 box over a streamed 9x9 raw patch.
// Compares UNSCALED 3x3 sums — argmax-equivalent to comparing sums*(1/9).
// Returns true if converged (best move is (0,0)).
__device__ __forceinline__ bool step_interior(const float* __restrict__ img, int& x, int& y) {
  const float* base = img + (y - 4) * kW + (x - 4);
  float r[3][9];
#pragma unroll
  for (int b = 0; b < 9; ++b) r[0][b] = base[b];           // row y-4
#pragma unroll
  for (int b = 0; b < 9; ++b) r[1][b] = base[kW + b];      // row y-3

  float best = -__builtin_inff();
  int bdy = 0, bdx = 0;
#pragma unroll
  for (int a = 1; a <= 7; ++a) {                           // center row y-4+a (dy = a-4)
    float* rt = r[(a - 1) % 3];
    float* rm = r[a % 3];
    float* rb = r[(a + 1) % 3];
#pragma unroll
    for (int b = 0; b < 9; ++b) rb[b] = base[(a + 1) * kW + b];
    float cs[9];
#pragma unroll
    for (int b = 0; b < 9; ++b) cs[b] = rt[b] + rm[b] + rb[b];
#pragma unroll
    for (int bp = 1; bp <= 7; ++bp) {                      // center col x-4+bp (dx = bp-4)
      float v = cs[bp - 1] + cs[bp] + cs[bp + 1];
      if (v > best) { best = v; bdy = a - 4; bdx = bp - 4; } // strict >: first-max (dy-major)
    }
  }
  if (bdy == 0 && bdx == 0) return true;
  x += bdx; y += bdy;
  return false;
}

// Border-safe gravity step: exact reference semantics (clip coords, zero-padded
// smooth, dy-major first-max tie-break on the scaled values).
__device__ __forceinline__ bool step_border(const float* __restrict__ img, int& x, int& y) {
  float best = -__builtin_inff();
  int bx = x, by = y;
#pragma unroll 1
  for (int dy = -kRadius; dy <= kRadius; ++dy) {
    int yy = min(max(y + dy, 0), kH - 1);
#pragma unroll 1
    for (int dx = -kRadius; dx <= kRadius; ++dx) {
      int xx = min(max(x + dx, 0), kW - 1);
      float v = smooth_at(img, yy, xx);
      if (v > best) { best = v; bx = xx; by = yy; }
    }
  }
  if (bx == x && by == y) return true;
  x = bx; y = by;
  return false;
}

// One block per batch image, one thread per point. Fused:
//   gravity hill-climb -> WMMA pairwise-distance tiles -> overlap resolve -> peak gather.
__global__ __launch_bounds__(kN) void ggd_fused_kernel(
    const float* __restrict__ depth,   // (B,1,H,W) f32
    const int* __restrict__ pts,       // (B,N,2) i32  (x,y)
    float* __restrict__ out)           // [B*N*2] end (as float) then [B*N] peak
{
  extern __shared__ float smem[];
  float* Se   = smem;                    // 256*65: dist^2(end_i, end_j) strip (padded)
  float* Sd   = Se + kN * kStripPad;     // 256*65: dist^2(end_i, start_j) strip (padded)
  float* exd  = Sd + kN * kStripPad;     // 256: end x (float)
  float* eyd  = exd + kN;                // 256: end y
  float* sxd  = eyd + kN;                // 256: start x
  float* syd  = sxd + kN;                // 256: start y
  float* ne   = syd + kN;                // 256: |end|^2
  float* ns   = ne + kN;                 // 256: |start|^2
  int*   iex  = (int*)(ns + kN);         // 256: end x (int)
  int*   iey  = iex + kN;                // 256: end y (int)
  int*   ldrA = iey + kN;                // 256: leader
  int*   wnrA = ldrA + kN;               // 256: winner
  int*   hasA = wnrA + kN;               // 256: has-overlap flag

  const int bi = blockIdx.x;
  const int i  = threadIdx.x;            // point index within batch
  const float* __restrict__ img = depth + (size_t)bi * kH * kW;

  // ---------------- gravity hill-climb (per point) ----------------
  const int sx0 = pts[(bi * kN + i) * 2 + 0];
  const int sy0 = pts[(bi * kN + i) * 2 + 1];
  int x = sx0, y = sy0;
#pragma unroll 1
  for (int it = 0; it < kMaxIters; ++it) {
    bool conv;
    if (x >= 4 && x <= kW - 5 && y >= 4 && y <= kH - 5)
      conv = step_interior(img, x, y);   // fast separable path (no clipping possible)
    else
      conv = step_border(img, x, y);     // exact clipped/zero-padded path
    if (conv) break;                     // fixed point: identical to global-disp loop
  }

  // stage coordinates into LDS for the WMMA phase
  {
    float fx = (float)x, fy = (float)y;
    float gx = (float)sx0, gy = (float)sy0;
    exd[i] = fx; eyd[i] = fy;
    sxd[i] = gx; syd[i] = gy;
    ne[i] = fx * fx + fy * fy;
    ns[i] = gx * gx + gy * gy;
    iex[i] = x; iey[i] = y;
  }
  __syncthreads();

  // ---------------- pairwise distances via V_WMMA_F32_16X16X4_F32 ----------------
  // dist^2(i,j) = |a_i|^2 + |b_j|^2 - 2*(a_i . b_j); all values are small ints -> exact in f32.
  const int wave  = threadIdx.x >> 5;
  const int lane  = threadIdx.x & 31;
  const int lrow  = lane & 15;
  const int rhalf = (lane >> 4) << 3;   // 0 for lanes 0-15, 8 for lanes 16-31

  int   cnt    = 0;                 // row popcount of ov (includes diagonal)
  int   leader = -1;                // first j with ov[i][j]
  int   winner = -1;                // masked argmin_j dist^2(start_j, end_i), first-min
  float wmin   = __builtin_inff();

#pragma unroll 1
  for (int j0 = 0; j0 < kN; j0 += kStrip) {
    // 8 waves x 8 tiles cover the 256 x 64 strip with 16x16 WMMA tiles
#pragma unroll 1
    for (int t = wave * 8; t < wave * 8 + 8; ++t) {
      const int i0 = (t >> 2) << 4;                 // tile row base (0..240)
      const int jt = j0 + ((t & 3) << 4);           // tile col base within full matrix
      const bool lo = (lane < 16);
      // A: 16x4 (rows = end_i), K=0..1 hold (x,y), K=2..3 zero (lanes 16-31)
      v2f a;  a.x  = lo ? exd[i0 + lrow] : 0.0f;  a.y  = lo ? eyd[i0 + lrow] : 0.0f;
      // B: 4x16 (cols = end_j / start_j)
      v2f be; be.x = lo ? exd[jt + lrow] : 0.0f;  be.y = lo ? eyd[jt + lrow] : 0.0f;
      v2f bs; bs.x = lo ? sxd[jt + lrow] : 0.0f;  bs.y = lo ? syd[jt + lrow] : 0.0f;
      v8f c = {0.f, 0.f, 0.f, 0.f, 0.f, 0.f, 0.f, 0.f};
      v8f dE = __builtin_amdgcn_wmma_f32_16x16x4_f32(false, a, false, be, (short)0, c, false, false);
      v8f dS = __builtin_amdgcn_wmma_f32_16x16x4_f32(false, a, false, bs, (short)0, c, false, false);
      const int col = jt + lrow;
#pragma unroll
      for (int v = 0; v < 8; ++v) {
        const int row = i0 + v + rhalf;
        Se[row * kStripPad + (col - j0)] = ne[row] + ne[col] - 2.0f * dE[v];
        Sd[row * kStripPad + (col - j0)] = ne[row] + ns[col] - 2.0f * dS[v];
      }
    }
    __syncthreads();

    // ordered left-to-right row scan: running reductions match jnp.argmax/argmin tie rules
#pragma unroll 1
    for (int jj = 0; jj < kStrip; ++jj) {
      const int j = j0 + jj;
      const float de = Se[i * kStripPad + jj];
      if (de < 4.0f) {                       // dist < 2.0  <=>  dist^2 < 4.0 (exact)
        ++cnt;
        if (leader < 0) leader = j;
        const float dd = Sd[i * kStripPad + jj];
        if (dd < wmin) { wmin = dd; winner = j; }   // strict <: first-min tie-break
      }
    }
    __syncthreads();   // strip buffers reused next iteration
  }

  ldrA[i] = leader;
  wnrA[i] = winner;
  hasA[i] = (cnt > 1) ? 1 : 0;
  __syncthreads();

  // ---------------- resolve + peak gather ----------------
  const int L   = leader;            // always >= 0 (diagonal overlaps itself)
  const int wol = wnrA[L];           // winner of my leader
  int ox, oy;
  if (hasA[i]) {
    if (i == wol) { ox = iex[L]; oy = iey[L]; }   // keep: take leader's end
    else          { ox = sx0;    oy = sy0;    }   // bounce back to start
  } else          { ox = x;      oy = y;      }   // no overlap: keep own end

  const float pk = smooth_at(img, oy, ox);

  out[(bi * kN + i) * 2 + 0] = (float)ox;
  out[(bi * kN + i) * 2 + 1] = (float)oy;
  out[kB * kN * 2 + bi * kN + i] = pk;
}

extern "C" void kernel_launch(void* const* d_in, const int* in_sizes, int n_in,
                              void* d_out, int out_size, void* d_ws, size_t ws_size,
                              hipStream_t stream) {
  (void)in_sizes; (void)n_in; (void)out_size; (void)d_ws; (void)ws_size;
  const float* depth = (const float*)d_in[0];
  const int*   points = (const int*)d_in[1];
  float*       out = (float*)d_out;

  const size_t smem_bytes =
      (size_t)(2 * kN * kStripPad  // Se + Sd strips (bank-conflict padded)
               + 8 * kN            // exd, eyd, sxd, syd, ne, ns + iex, iey
               + 3 * kN)           // ldrA, wnrA, hasA
      * sizeof(float);

  // CDNA5 allows up to 320KB LDS per workgroup; opt in for >64KB dynamic LDS.
  (void)hipFuncSetAttribute((const void*)ggd_fused_kernel,
                            hipFuncAttributeMaxDynamicSharedMemorySize,
                            (int)smem_bytes);

  ggd_fused_kernel<<<kB, kN, smem_bytes, stream>>>(depth, points, out);
}